// RecSysGNN_25142738551261
// MI455X (gfx1250) — compile-verified
//
#include <hip/hip_runtime.h>
#include <hip/hip_bf16.h>

typedef float v2f __attribute__((ext_vector_type(2)));
typedef float v8f __attribute__((ext_vector_type(8)));

#define N_NODES 100000
#define EMB_DIM 64
#define NELEM   (N_NODES * EMB_DIM)   // 6,400,000 floats per node-feature buffer

// ---------------- helpers ----------------

__global__ void k_zero4(float4* __restrict__ p, int n4) {
  int i = blockIdx.x * blockDim.x + threadIdx.x;
  if (i < n4) p[i] = make_float4(0.f, 0.f, 0.f, 0.f);
}

// deg[dst] += 1.0 for every edge (float to match segment_sum of ones)
__global__ void k_deg(const int* __restrict__ dst, float* __restrict__ deg, int E) {
  int e = blockIdx.x * blockDim.x + threadIdx.x;
  if (e < E) atomicAdd(&deg[dst[e]], 1.0f);
}

// deg -> rsqrt(deg) in place (0 where deg==0)
__global__ void k_dinv(float* __restrict__ deg, int n) {
  int i = blockIdx.x * blockDim.x + threadIdx.x;
  if (i < n) {
    float d = deg[i];
    deg[i] = (d > 0.f) ? rsqrtf(d) : 0.f;
  }
}

// coef[e] = dinv[src]*dinv[dst]*exp(scale*attr)
__global__ void k_coef(const float* __restrict__ attr, const float* __restrict__ scale,
                       const int* __restrict__ src, const int* __restrict__ dst,
                       const float* __restrict__ dinv, float* __restrict__ coef, int E) {
  int e = blockIdx.x * blockDim.x + threadIdx.x;
  if (e < E) coef[e] = dinv[src[e]] * dinv[dst[e]] * expf(scale[0] * attr[e]);
}

// softmax over the 4 attention weights
__global__ void k_attn(const float* __restrict__ aw, float* __restrict__ attn) {
  if (threadIdx.x == 0 && blockIdx.x == 0) {
    float m = aw[0];
    for (int i = 1; i < 4; ++i) m = fmaxf(m, aw[i]);
    float e[4], s = 0.f;
    for (int i = 0; i < 4; ++i) { e[i] = expf(aw[i] - m); s += e[i]; }
    float inv = 1.f / s;
    for (int i = 0; i < 4; ++i) attn[i] = e[i] * inv;
  }
}

// out1 = emb0 ; out2 = attn[0]*emb0   (vectorized float4)
__global__ void k_init(const float4* __restrict__ emb, const float* __restrict__ attn,
                       float4* __restrict__ o1, float4* __restrict__ o2, int n4) {
  int i = blockIdx.x * blockDim.x + threadIdx.x;
  if (i < n4) {
    float a0 = attn[0];
    float4 v = emb[i];
    o1[i] = v;
    float4 w; w.x = a0 * v.x; w.y = a0 * v.y; w.z = a0 * v.z; w.w = a0 * v.w;
    o2[i] = w;
  }
}

// One wave per edge: lane l moves dims 2l,2l+1. Coalesced 256B gather from
// x[src], f32 atomic scatter-add into xn[dst]. L2-resident (25.6MB table).
__global__ void k_scatter(const float* __restrict__ x, const float* __restrict__ coef,
                          const int* __restrict__ src, const int* __restrict__ dst,
                          float* __restrict__ xn, int E) {
  int t = blockIdx.x * blockDim.x + threadIdx.x;
  int e = t >> 5;
  int lane = t & 31;
  if (e >= E) return;
  int s = src[e];
  int d = dst[e];
  float c = coef[e];
  float2 v = ((const float2*)(x + s * EMB_DIM))[lane];
  float* o = xn + d * EMB_DIM + lane * 2;
  atomicAdd(o,     v.x * c);
  atomicAdd(o + 1, v.y * c);
}

// out[n] += attn[lidx] * x[n], done via V_WMMA_F32_16X16X4_F32 (exact f32 FMA).
// Per wave: 16 contiguous elements in the N dimension.
//   A[m,k] = attn  at k==0 else 0  -> a[0]=attn for lanes 0-15 (K=0 slot), rest 0
//   B[0,n] = x[base+n]             -> b[0]=x for lanes 0-15 (K=0 row), rest 0
//   C[m,n] = out[base+n] broadcast -> every c[i] = out[base + (lane&15)]
//   D[0,n] read from d[0], lanes 0-15 (M=0, N=lane)
__global__ void k_combine(const float* __restrict__ x, const float* __restrict__ attn,
                          int lidx, float* __restrict__ out, int n) {
  int t = blockIdx.x * blockDim.x + threadIdx.x;
  int wave = t >> 5;
  int lane = t & 31;
  int base = wave * 16;
  if (base >= n) return;                 // uniform per wave; EXEC stays all-ones
  float al = attn[lidx];
  int col = lane & 15;
  float xv = x[base + col];
  float ov = out[base + col];
  v2f a; a[0] = (lane < 16) ? al : 0.0f; a[1] = 0.0f;
  v2f b; b[0] = (lane < 16) ? xv : 0.0f; b[1] = 0.0f;
  v8f c;
#pragma unroll
  for (int i = 0; i < 8; ++i) c[i] = ov;
  v8f d = __builtin_amdgcn_wmma_f32_16x16x4_f32(false, a, false, b, (short)0, c,
                                                false, false);
  if (lane < 16) out[base + lane] = d[0];
}

// ---------------- launcher ----------------

extern "C" void kernel_launch(void* const* d_in, const int* in_sizes, int n_in,
                              void* d_out, int out_size, void* d_ws, size_t ws_size,
                              hipStream_t stream) {
  const float* emb   = (const float*)d_in[0];   // [N_NODES, 64]
  const float* aw    = (const float*)d_in[1];   // [4]
  const float* scale = (const float*)d_in[2];   // [1]
  const float* attr  = (const float*)d_in[3];   // [E]
  const int*   eidx  = (const int*)d_in[4];     // [2, E]
  const int E = in_sizes[3];
  const int* src = eidx;
  const int* dst = eidx + E;

  float* out1 = (float*)d_out;          // emb0 copy
  float* out2 = out1 + NELEM;           // attention-combined output (accumulator)

  // workspace layout (floats), all chunk starts 16B-aligned
  float* W    = (float*)d_ws;
  float* dinv = W;                      // 100,000 (zeroed as 100,352 via float4)
  float* attn = W + 100352;             // 4
  float* coef = W + 100384;             // E = 3,200,000
  float* xA   = coef + E;               // NELEM
  float* xB   = xA + NELEM;             // NELEM

  const int BS = 256;
  const int n4      = NELEM / 4;                 // 1,600,000
  const int gZeroX  = (n4 + BS - 1) / BS;
  const int gZeroD  = (100352 / 4 + BS - 1) / BS;
  const int gEdge   = (E + BS - 1) / BS;
  const int gNode   = (N_NODES + BS - 1) / BS;
  const int gScat   = (int)(((long)E * 32 + BS - 1) / BS);     // wave per edge
  const int gComb   = (int)(((long)(NELEM / 16) * 32 + BS - 1) / BS);

  // 1) degree -> rsqrt
  k_zero4<<<gZeroD, BS, 0, stream>>>((float4*)dinv, 100352 / 4);
  k_deg  <<<gEdge,  BS, 0, stream>>>(dst, dinv, E);
  k_dinv <<<gNode,  BS, 0, stream>>>(dinv, N_NODES);

  // 2) per-edge coefficient, attention softmax
  k_coef <<<gEdge, BS, 0, stream>>>(attr, scale, src, dst, dinv, coef, E);
  k_attn <<<1, 32, 0, stream>>>(aw, attn);

  // 3) out1 = emb0 ; out2 = attn0*emb0
  k_init <<<gZeroX, BS, 0, stream>>>((const float4*)emb, attn,
                                     (float4*)out1, (float4*)out2, n4);

  // 4) three propagation layers, ping-pong xA/xB; accumulate into out2 via WMMA
  // layer 1: emb -> xA
  k_zero4   <<<gZeroX, BS, 0, stream>>>((float4*)xA, n4);
  k_scatter <<<gScat,  BS, 0, stream>>>(emb, coef, src, dst, xA, E);
  k_combine <<<gComb,  BS, 0, stream>>>(xA, attn, 1, out2, NELEM);
  // layer 2: xA -> xB
  k_zero4   <<<gZeroX, BS, 0, stream>>>((float4*)xB, n4);
  k_scatter <<<gScat,  BS, 0, stream>>>(xA, coef, src, dst, xB, E);
  k_combine <<<gComb,  BS, 0, stream>>>(xB, attn, 2, out2, NELEM);
  // layer 3: xB -> xA
  k_zero4   <<<gZeroX, BS, 0, stream>>>((float4*)xA, n4);
  k_scatter <<<gScat,  BS, 0, stream>>>(xB, coef, src, dst, xA, E);
  k_combine <<<gComb,  BS, 0, stream>>>(xA, attn, 3, out2, NELEM);
}